// YOLOv3Layer_13383118094575
// MI455X (gfx1250) — compile-verified
//
#include <hip/hip_runtime.h>
#include <math.h>
#include <float.h>

#define N_CLASSES 80
#define B_SZ 16
#define TOPK 100
#define MAX_PER_IMAGE 50
#define SCORE_THRESH 0.3f
#define NMS_THRESH 0.3f

#define G0 19
#define G1 38
#define G2 76
#define S0 (G0*G0*3)        /* 1083  */
#define S1 (G1*G1*3)        /* 4332  */
#define S2 (G2*G2*3)        /* 17328 */
#define P_TOT (S0+S1+S2)    /* 22743 */
#define INPUT_SZ 608.0f

typedef __attribute__((ext_vector_type(2))) float v2f;
typedef __attribute__((ext_vector_type(8))) float v8f;

__device__ __forceinline__ float sigmoidf(float x) { return 1.0f / (1.0f + expf(-x)); }

// ---------------------------------------------------------------------------
// Kernel 1: decode all scales. One thread per (batch, prediction).
// Writes boxes (B,P,4) and scores transposed to (B,80,P) in workspace.
// ---------------------------------------------------------------------------
__global__ __launch_bounds__(256) void decode_kernel(
    const float* __restrict__ f0, const float* __restrict__ f1,
    const float* __restrict__ f2, const float* __restrict__ anchors,
    const float* __restrict__ image_shape,
    float* __restrict__ boxes, float* __restrict__ scores)
{
    int gid = blockIdx.x * blockDim.x + threadIdx.x;
    if (gid >= B_SZ * P_TOT) return;
    int b = gid / P_TOT;
    int p = gid - b * P_TOT;

    const float* feat; int gsz, pl, scale;
    if (p < S0)           { feat = f0; gsz = G0; pl = p;            scale = 0; }
    else if (p < S0 + S1) { feat = f1; gsz = G1; pl = p - S0;       scale = 1; }
    else                  { feat = f2; gsz = G2; pl = p - S0 - S1;  scale = 2; }

    int a    = pl % 3;
    int cell = pl / 3;
    int gy   = cell / gsz;
    int gx   = cell - gy * gsz;
    int ai   = (2 - scale) * 3 + a;          // ANCHOR_MASK

    const size_t plane = (size_t)gsz * gsz;
    const float* base = feat + ((size_t)b * 255 + (size_t)a * 85) * plane
                             + (size_t)gy * gsz + gx;
    __builtin_prefetch(base + 5 * plane, 0, 1);   // global_prefetch_b8

    float t0 = base[0];
    float t1 = base[plane];
    float t2 = base[2 * plane];
    float t3 = base[3 * plane];
    float t4 = base[4 * plane];

    float img_h = image_shape[0], img_w = image_shape[1];
    float r     = fminf(INPUT_SZ / img_h, INPUT_SZ / img_w);
    float new_h = roundf(img_h * r), new_w = roundf(img_w * r);
    float off_h = (INPUT_SZ - new_h) * 0.5f / INPUT_SZ;
    float off_w = (INPUT_SZ - new_w) * 0.5f / INPUT_SZ;
    float scl_h = INPUT_SZ / new_h, scl_w = INPUT_SZ / new_w;

    float xg = (sigmoidf(t0) + (float)gx) / (float)gsz;
    float yg = (sigmoidf(t1) + (float)gy) / (float)gsz;
    float aw = anchors[ai * 2 + 0], ah = anchors[ai * 2 + 1];
    float wn = aw * expf(t2) / INPUT_SZ;
    float hn = ah * expf(t3) / INPUT_SZ;

    float yc = (yg - off_h) * scl_h;
    float xc = (xg - off_w) * scl_w;
    float hc = hn * scl_h, wc = wn * scl_w;

    float y1 = (yc - 0.5f * hc) * img_h;
    float x1 = (xc - 0.5f * wc) * img_w;
    float y2 = (yc + 0.5f * hc) * img_h;
    float x2 = (xc + 0.5f * wc) * img_w;

    float* bo = boxes + ((size_t)b * P_TOT + p) * 4;
    bo[0] = y1; bo[1] = x1; bo[2] = y2; bo[3] = x2;

    float conf = sigmoidf(t4);
    for (int c = 0; c < N_CLASSES; ++c) {
        float s = conf * sigmoidf(base[(size_t)(5 + c) * plane]);
        scores[((size_t)b * N_CLASSES + c) * P_TOT + p] = s;
    }
}

// ---------------------------------------------------------------------------
// Kernel 2: per-(batch,class) top-100 + greedy NMS. One 256-thread block each.
// area[i]+area[j] outer-sum (IoU denominator) computed with WMMA f32 16x16x4.
// ---------------------------------------------------------------------------
__global__ __launch_bounds__(256) void nms_kernel(
    const float* __restrict__ boxes, const float* __restrict__ scores,
    float* __restrict__ out)
{
    const int tid = threadIdx.x;
    const int b   = blockIdx.x / N_CLASSES;
    const int c   = blockIdx.x % N_CLASSES;

    __shared__ float rv[256];
    __shared__ int   ri[256];
    __shared__ float selV[TOPK];
    __shared__ int   selI[TOPK];
    __shared__ float bxs[112 * 4];
    __shared__ float area[112];
    __shared__ float areaSum[112 * 112];
    __shared__ unsigned char keepA[112];

    const float* sc = scores + ((size_t)b * N_CLASSES + c) * P_TOT;
    const int chunk = (P_TOT + 255) / 256;                 // 89 elements/thread
    int beg = tid * chunk;
    int end = min(beg + chunk, P_TOT);
    unsigned selMask[3] = {0u, 0u, 0u};                    // 89 bits

    // ---- iterative argmax top-100 (thresholded; ties -> lowest index) ----
    for (int it = 0; it < TOPK; ++it) {
        float bv = -FLT_MAX; int bi = 0x7fffffff;
        int li = 0;
        for (int i = beg; i < end; ++i, ++li) {
            if ((selMask[li >> 5] >> (li & 31)) & 1u) continue;
            float v = sc[i];
            v = (v > SCORE_THRESH) ? v : -1.0f;
            if (v > bv) { bv = v; bi = i; }
        }
        rv[tid] = bv; ri[tid] = bi;
        __syncthreads();
        for (int s = 128; s > 0; s >>= 1) {
            if (tid < s) {
                float v2 = rv[tid + s]; int i2 = ri[tid + s];
                if (v2 > rv[tid] || (v2 == rv[tid] && i2 < ri[tid])) {
                    rv[tid] = v2; ri[tid] = i2;
                }
            }
            __syncthreads();
        }
        int w = ri[0];
        if (tid == 0) { selV[it] = rv[0]; selI[it] = w; }
        if (w >= beg && w < end) {
            int l = w - beg;
            selMask[l >> 5] |= (1u << (l & 31));
        }
        __syncthreads();
    }

    // ---- gather selected boxes, areas (padded to 112 with zeros) ----
    for (int k = tid; k < 112; k += 256) {
        if (k < TOPK) {
            const float* bp = boxes + ((size_t)b * P_TOT + selI[k]) * 4;
            float y1 = bp[0], x1 = bp[1], y2 = bp[2], x2 = bp[3];
            bxs[k * 4 + 0] = y1; bxs[k * 4 + 1] = x1;
            bxs[k * 4 + 2] = y2; bxs[k * 4 + 3] = x2;
            area[k]  = (y2 - y1) * (x2 - x1);
            keepA[k] = (selV[k] > SCORE_THRESH) ? 1 : 0;
        } else {
            bxs[k * 4 + 0] = 0.f; bxs[k * 4 + 1] = 0.f;
            bxs[k * 4 + 2] = 0.f; bxs[k * 4 + 3] = 0.f;
            area[k] = 0.f; keepA[k] = 0;
        }
    }
    __syncthreads();

    // ---- areaSum[i][j] = area[i] + area[j] via V_WMMA_F32_16X16X4_F32 ----
    // A (16x4) row m = [area_m, 1, 0, 0]; B (4x16) col n = [1, area_n, 0, 0]^T
    // D[m][n] = area_m*1 + 1*area_n (exact in f32). 7x7 tiles over 112x112,
    // tiles distributed wave-uniformly across the block's 8 waves so EXEC is
    // all-ones at each WMMA.
    {
        int wave = tid >> 5, lane = tid & 31;
        for (int t = wave; t < 49; t += 8) {
            int tm = t / 7, tn = t % 7;
            bool lo = lane < 16;        // lanes 0-15: K=0/1; lanes 16-31: K=2/3 (zeros)
            int  l15 = lane & 15;
            v2f A, Bm;
            A[0]  = lo ? area[tm * 16 + l15] : 0.f;  // K=0
            A[1]  = lo ? 1.0f                : 0.f;  // K=1
            Bm[0] = lo ? 1.0f                : 0.f;  // B[0][n]
            Bm[1] = lo ? area[tn * 16 + l15] : 0.f;  // B[1][n]
            v8f C = {};
            v8f D = __builtin_amdgcn_wmma_f32_16x16x4_f32(
                false, A, false, Bm, (short)0, C, false, false);
            int col = tn * 16 + l15;
            int rb  = tm * 16 + (lo ? 0 : 8);        // C/D layout: lanes16-31 -> M=8..15
            for (int rr = 0; rr < 8; ++rr)
                areaSum[(rb + rr) * 112 + col] = D[rr];
        }
    }
    __syncthreads();

    // ---- greedy sequential suppression (thread j handles column j) ----
    int j = tid;
    for (int i = 0; i < TOPK; ++i) {
        if (keepA[i] && j < TOPK && j > i) {
            float iy1 = fmaxf(bxs[i * 4 + 0], bxs[j * 4 + 0]);
            float ix1 = fmaxf(bxs[i * 4 + 1], bxs[j * 4 + 1]);
            float iy2 = fminf(bxs[i * 4 + 2], bxs[j * 4 + 2]);
            float ix2 = fminf(bxs[i * 4 + 3], bxs[j * 4 + 3]);
            float inter = fmaxf(iy2 - iy1, 0.f) * fmaxf(ix2 - ix1, 0.f);
            float iou = inter / (areaSum[i * 112 + j] - inter + 1e-9f);
            if (iou > NMS_THRESH) keepA[j] = 0;
        }
        __syncthreads();
    }

    // ---- emit dets * keep ----
    if (tid < TOPK) {
        float k = keepA[tid] ? 1.0f : 0.0f;
        float* o = out + (((size_t)b * N_CLASSES + c) * TOPK + tid) * 5;
        o[0] = bxs[tid * 4 + 0] * k;
        o[1] = bxs[tid * 4 + 1] * k;
        o[2] = bxs[tid * 4 + 2] * k;
        o[3] = bxs[tid * 4 + 3] * k;
        o[4] = selV[tid] * k;
    }
}

// ---------------------------------------------------------------------------
// Kernel 3: per-image cap to top-50 detections over the 8000 (class,slot)
// scores; zero everything outside the mask (in-place on d_out).
// ---------------------------------------------------------------------------
__global__ __launch_bounds__(256) void cap_kernel(float* __restrict__ out)
{
    const int tid = threadIdx.x;
    const int b   = blockIdx.x;
    float* d = out + (size_t)b * N_CLASSES * TOPK * 5;
    const int NF = N_CLASSES * TOPK;                  // 8000

    __shared__ float rv[256];
    __shared__ int   ri[256];
    __shared__ unsigned char keepM[N_CLASSES * TOPK];

    for (int f = tid; f < NF; f += 256) keepM[f] = 0;
    __syncthreads();

    const int chunk = (NF + 255) / 256;               // 32
    int beg = tid * chunk;
    int end = min(beg + chunk, NF);
    unsigned selMask = 0u;

    for (int it = 0; it < MAX_PER_IMAGE; ++it) {
        float bv = -FLT_MAX; int bi = 0x7fffffff;
        int li = 0;
        for (int f = beg; f < end; ++f, ++li) {
            if ((selMask >> li) & 1u) continue;
            float v = d[f * 5 + 4];
            if (v > bv) { bv = v; bi = f; }
        }
        rv[tid] = bv; ri[tid] = bi;
        __syncthreads();
        for (int s = 128; s > 0; s >>= 1) {
            if (tid < s) {
                float v2 = rv[tid + s]; int i2 = ri[tid + s];
                if (v2 > rv[tid] || (v2 == rv[tid] && i2 < ri[tid])) {
                    rv[tid] = v2; ri[tid] = i2;
                }
            }
            __syncthreads();
        }
        int w = ri[0];
        if (tid == 0) keepM[w] = 1;
        if (w >= beg && w < end) selMask |= (1u << (w - beg));
        __syncthreads();
    }

    for (int f = tid; f < NF; f += 256) {
        if (!keepM[f]) {
            d[f * 5 + 0] = 0.f; d[f * 5 + 1] = 0.f; d[f * 5 + 2] = 0.f;
            d[f * 5 + 3] = 0.f; d[f * 5 + 4] = 0.f;
        }
    }
}

// ---------------------------------------------------------------------------
extern "C" void kernel_launch(void* const* d_in, const int* in_sizes, int n_in,
                              void* d_out, int out_size, void* d_ws, size_t ws_size,
                              hipStream_t stream)
{
    (void)in_sizes; (void)n_in; (void)out_size; (void)ws_size;

    const float* f0          = (const float*)d_in[0];
    const float* f1          = (const float*)d_in[1];
    const float* f2          = (const float*)d_in[2];
    const float* anchors     = (const float*)d_in[3];
    const float* image_shape = (const float*)d_in[4];
    float* out = (float*)d_out;

    // Workspace layout: boxes (B,P,4) then scores (B,80,P), 256B-aligned.
    float* ws_boxes = (float*)d_ws;
    size_t boxes_bytes = (size_t)B_SZ * P_TOT * 4 * sizeof(float);
    size_t off = (boxes_bytes + 255) & ~((size_t)255);
    float* ws_scores = (float*)((char*)d_ws + off);

    int total = B_SZ * P_TOT;
    decode_kernel<<<(total + 255) / 256, 256, 0, stream>>>(
        f0, f1, f2, anchors, image_shape, ws_boxes, ws_scores);
    nms_kernel<<<B_SZ * N_CLASSES, 256, 0, stream>>>(ws_boxes, ws_scores, out);
    cap_kernel<<<B_SZ, 256, 0, stream>>>(out);
}